// BilinearPoolingFusion_82446192214447
// MI455X (gfx1250) — compile-verified
//
#include <hip/hip_runtime.h>
#include <hip/hip_bf16.h>

typedef __attribute__((ext_vector_type(2))) float v2f;
typedef __attribute__((ext_vector_type(8))) float v8f;

#define D 512

// ---------------- Kernel 1: partial column sums of x2 ----------------
// grid: 8 batches * 16 chunks of 64 rows; 512 threads (one per j)
__global__ __launch_bounds__(512) void k_colsum(const float* __restrict__ x2,
                                                float* __restrict__ part) {
    const int b = blockIdx.x >> 4;
    const int c = blockIdx.x & 15;
    const int j = threadIdx.x;
    const float* p = x2 + ((size_t)b * 1024 + (size_t)c * 64) * D + j;
    float s = 0.0f;
#pragma unroll 8
    for (int l = 0; l < 64; ++l) s += p[(size_t)l * D];
    part[(size_t)blockIdx.x * D + j] = s;
}

// grid: 8 blocks (batch), 512 threads (j). Writes x2m transposed: [j][b].
__global__ __launch_bounds__(512) void k_colsum_fin(const float* __restrict__ part,
                                                    float* __restrict__ x2mT) {
    const int b = blockIdx.x;
    const int j = threadIdx.x;
    float s = 0.0f;
#pragma unroll
    for (int c = 0; c < 16; ++c) s += part[((size_t)(b * 16 + c)) * D + j];
    x2mT[j * 8 + b] = s * (1.0f / 1024.0f);
}

// ---------------- Kernel 2: Tt[b][i][o] = sum_j W[o][i][j] * x2m[b][j] ------
// GEMM rows r = i*512 + o  (A row r = W[o][i][:], contiguous K=512),
// N = batch (8 used, padded to 16).  f32 WMMA 16x16x4, K-loop over 512.
// One wave per 16-row tile; 8 waves / block.
__global__ __launch_bounds__(256) void k_wtimes(const float* __restrict__ W,
                                                const float* __restrict__ x2mT,
                                                float* __restrict__ Tt) {
    __shared__ float sB[D * 16];        // [j][n] x2m padded to N=16 (32 KB)
    __shared__ float sOut[8][16 * 17];  // per-wave D-tile transpose staging

    const int tid = threadIdx.x;
    for (int idx = tid; idx < D * 16; idx += 256) {
        const int j = idx >> 4, n = idx & 15;
        sB[idx] = (n < 8) ? x2mT[j * 8 + n] : 0.0f;
    }
    __syncthreads();

    const int wave = tid >> 5;
    const int lane = tid & 31;
    const int hi   = lane >> 4;   // half-wave select
    const int ln   = lane & 15;

    const int rbase = (blockIdx.x * 8 + wave) * 16; // 16 consecutive (i,o) rows
    const int i     = rbase >> 9;                   // same i for whole tile
    const int obase = rbase & 511;
    // This lane's A row: M = ln  ->  r = rbase + ln  ->  W[o = obase+ln][i][:]
    const float* Arow = W + ((size_t)(obase + ln) * D + (size_t)i) * D;

    v8f acc = {0.f, 0.f, 0.f, 0.f, 0.f, 0.f, 0.f, 0.f};
#pragma unroll 4
    for (int kk = 0; kk < D; kk += 4) {
        const int k0 = kk + 2 * hi;
        // A fragment: 16x4 f32 layout (V0: K=k0, V1: K=k0+1) — stream W with NT hint
        v2f a = __builtin_nontemporal_load((const v2f*)(Arow + k0));
        // B fragment: 4x16 f32 layout (V0: K=k0, V1: K=k0+1 at N=ln)
        v2f b;
        b.x = sB[k0 * 16 + ln];
        b.y = sB[(k0 + 1) * 16 + ln];
        acc = __builtin_amdgcn_wmma_f32_16x16x4_f32(false, a, false, b,
                                                    (short)0, acc, false, false);
    }

    // Transpose D tile through LDS so each batch plane gets coalesced stores.
    float* st = &sOut[wave][0];
#pragma unroll
    for (int v = 0; v < 8; ++v) st[(v + 8 * hi) * 17 + ln] = acc[v];
    __syncthreads();
#pragma unroll
    for (int e = 0; e < 4; ++e) {
        const int linear = e * 32 + lane;  // 0..127 => n=0..7, m=0..15
        const int n = linear >> 4;
        const int m = linear & 15;
        Tt[(size_t)n * (D * D) + (size_t)(rbase + m)] = st[m * 17 + n];
    }
}

// ---------------- Kernel 3: out[b] = x1[b] @ Tt[b] + bias -------------------
// M=1024, N=512, K=512 per batch. Block = 8 waves sharing one 16-col B slice
// staged in LDS; each wave owns one 16x16 C tile.
__global__ __launch_bounds__(256) void k_gemm(const float* __restrict__ x1,
                                              const float* __restrict__ Tt,
                                              const float* __restrict__ bias,
                                              float* __restrict__ out) {
    __shared__ float sB[D * 16]; // [k][n] slice of Tt[b][:, nt*16 .. +16) (32 KB)

    const int tid = threadIdx.x;
    const int nt  = blockIdx.x & 31;         // N tile (512/16)
    const int mg  = (blockIdx.x >> 5) & 7;   // M group (8 waves * 16 rows)
    const int b   = blockIdx.x >> 8;         // batch

    const float* Bmat = Tt + (size_t)b * D * D + (size_t)nt * 16;
    for (int idx = tid; idx < D * 16; idx += 256) {
        const int k = idx >> 4, n = idx & 15;
        sB[idx] = Bmat[(size_t)k * D + n];
    }
    __syncthreads();

    const int wave = tid >> 5;
    const int lane = tid & 31;
    const int hi   = lane >> 4;
    const int ln   = lane & 15;

    const int mbase = (mg * 8 + wave) * 16;
    const float* Arow = x1 + (size_t)b * 1024 * D + (size_t)(mbase + ln) * D;

    v8f acc = {0.f, 0.f, 0.f, 0.f, 0.f, 0.f, 0.f, 0.f};
#pragma unroll 4
    for (int kk = 0; kk < D; kk += 4) {
        const int k0 = kk + 2 * hi;
        v2f a = *(const v2f*)(Arow + k0);
        v2f bb;
        bb.x = sB[k0 * 16 + ln];
        bb.y = sB[(k0 + 1) * 16 + ln];
        acc = __builtin_amdgcn_wmma_f32_16x16x4_f32(false, a, false, bb,
                                                    (short)0, acc, false, false);
    }

    const float bs = bias[nt * 16 + ln];     // column-constant across the tile
    float* orow = out + (size_t)b * 1024 * D + (size_t)mbase * D
                      + (size_t)nt * 16 + ln;
#pragma unroll
    for (int v = 0; v < 8; ++v) {            // row = mbase + v + 8*hi, col fixed
        orow[(size_t)(v + 8 * hi) * D] = acc[v] + bs;
    }
}

// ---------------------------------------------------------------------------
extern "C" void kernel_launch(void* const* d_in, const int* in_sizes, int n_in,
                              void* d_out, int out_size, void* d_ws, size_t ws_size,
                              hipStream_t stream) {
    const float* x1   = (const float*)d_in[0];  // [8,1024,512]
    const float* x2   = (const float*)d_in[1];  // [8,1024,512]
    const float* W    = (const float*)d_in[2];  // [512,512,512]
    const float* bias = (const float*)d_in[3];  // [512]
    float* out = (float*)d_out;                 // [8,1024,512]

    char* ws = (char*)d_ws;
    float* Tt   = (float*)ws;                                   // 8 MiB  [8][512][512]
    float* x2mT = (float*)(ws + (size_t)8 * 1024 * 1024);       // 16 KiB [512][8]
    float* part = (float*)(ws + (size_t)8 * 1024 * 1024 + 32 * 1024); // 256 KiB

    k_colsum    <<<128,  512, 0, stream>>>(x2, part);
    k_colsum_fin<<<8,    512, 0, stream>>>(part, x2mT);
    k_wtimes    <<<2048, 256, 0, stream>>>(W, x2mT, Tt);
    k_gemm      <<<2048, 256, 0, stream>>>(x1, Tt, bias, out);
}